// ConvGNN_39599598469674
// MI455X (gfx1250) — compile-verified
//
#include <hip/hip_runtime.h>
#include <hip/hip_bf16.h>

typedef __attribute__((ext_vector_type(2))) float v2f;
typedef __attribute__((ext_vector_type(8))) float v8f;

#define NNODES 50000
#define NEDGES 800000
#define NGRAPH 100
#define NPG    500
#define KTOP   250
#define DIM    64
#define NOUT   10

// ---------------------------------------------------------------- utilities
__global__ void set_val_kernel(float* __restrict__ p, float v, int n) {
  int i = blockIdx.x * blockDim.x + threadIdx.x;
  if (i < n) p[i] = v;
}

__global__ void deg_kernel(const int* __restrict__ col, float* __restrict__ deg, int ne) {
  int e = blockIdx.x * blockDim.x + threadIdx.x;
  if (e < ne) atomicAdd(&deg[col[e]], 1.0f);
}

__global__ void rsqrt_kernel(const float* __restrict__ deg, float* __restrict__ dis, int n) {
  int i = blockIdx.x * blockDim.x + threadIdx.x;
  if (i < n) dis[i] = rsqrtf(deg[i]);  // deg >= 1 always (self loops)
}

// ------------------------------------------------- H = X[N,64] @ W[64,64]
// One wave32 per 16-row tile. fp32 WMMA 16x16x4, 16 K-steps x 4 N-tiles.
__global__ void gemm_n64_wmma(const float* __restrict__ X, const float* __restrict__ W,
                              float* __restrict__ H, int nrows) {
  int gwave = (int)((blockIdx.x * blockDim.x + threadIdx.x) >> 5);
  int lane  = threadIdx.x & 31;
  int row0  = gwave << 4;
  if (row0 >= nrows) return;           // wave-uniform: EXEC stays all-1s
  int m  = lane & 15;                  // row (A) / col (B,C) within tile
  int kh = lane >> 4;                  // 0: K pair {k,k+1}; 1: {k+2,k+3}
  v8f acc[4] = {v8f{0.f}, v8f{0.f}, v8f{0.f}, v8f{0.f}};
  const float* xrow = X + (size_t)(row0 + m) * DIM;
  for (int k0 = 0; k0 < DIM; k0 += 4) {
    int ka = k0 + 2 * kh;
    v2f a;
    a.x = xrow[ka + 0];
    a.y = xrow[ka + 1];
#pragma unroll
    for (int nt = 0; nt < 4; ++nt) {
      const float* wp = W + (size_t)ka * DIM + nt * 16 + m;
      v2f b;
      b.x = wp[0];                     // row K=ka
      b.y = wp[DIM];                   // row K=ka+1
      acc[nt] = __builtin_amdgcn_wmma_f32_16x16x4_f32(
          false, a, false, b, (short)0, acc[nt], false, false);
    }
  }
#pragma unroll
  for (int nt = 0; nt < 4; ++nt) {
#pragma unroll
    for (int i = 0; i < 8; ++i) {
      int r = row0 + i + 8 * kh;       // C layout: vgpr i -> M=i / M=i+8
      H[(size_t)r * DIM + nt * 16 + m] = acc[nt][i];
    }
  }
}

// ------------------------------------- acc[col] += H[row] * dis[row]*dis[col]
// 16 threads per edge, float4 each. Memory-roofline kernel.
__global__ void edge_scatter_kernel(const int* __restrict__ row, const int* __restrict__ col,
                                    const float* __restrict__ dis, const float* __restrict__ H,
                                    float* __restrict__ acc, int ne) {
  int idx = blockIdx.x * blockDim.x + threadIdx.x;
  int e = idx >> 4;
  if (e >= ne) return;
  int q = (idx & 15) << 2;
  int r = row[e], c = col[e];
  float nrm = dis[r] * dis[c];
  float4 hv = *(const float4*)(H + (size_t)r * DIM + q);
  float* dst = acc + (size_t)c * DIM + q;
  atomicAdd(dst + 0, hv.x * nrm);
  atomicAdd(dst + 1, hv.y * nrm);
  atomicAdd(dst + 2, hv.z * nrm);
  atomicAdd(dst + 3, hv.w * nrm);
}

// ---------------------- y = relu(acc + h*dis^2 (self loop) + bias), in place
__global__ void finalize_kernel(float* __restrict__ acc, const float* __restrict__ H,
                                const float* __restrict__ dis, const float* __restrict__ bias,
                                int total) {
  int idx = blockIdx.x * blockDim.x + threadIdx.x;
  if (idx >= total) return;
  int i = idx >> 6, d = idx & 63;
  float s = dis[i];
  float v = acc[idx] + H[idx] * s * s + bias[d];
  acc[idx] = v > 0.f ? v : 0.f;
}

// ---------------------------------------------------------- pooling pieces
__global__ void wnorm_kernel(const float* __restrict__ pw, float* __restrict__ invw) {
  __shared__ float sh[64];
  int t = threadIdx.x;
  float v = pw[t];
  sh[t] = v * v;
  __syncthreads();
  for (int s = 32; s > 0; s >>= 1) {
    if (t < s) sh[t] += sh[t + s];
    __syncthreads();
  }
  if (t == 0) *invw = rsqrtf(sh[0]);
}

// score[n] = tanh(dot(x[n], w) / ||w||); one wave32 per node
__global__ void score_kernel(const float* __restrict__ X, const float* __restrict__ pw,
                             const float* __restrict__ invw, float* __restrict__ score, int n) {
  int gw = (int)((blockIdx.x * blockDim.x + threadIdx.x) >> 5);
  int lane = threadIdx.x & 31;
  if (gw >= n) return;
  const float* xr = X + (size_t)gw * DIM;
  float d = xr[lane] * pw[lane] + xr[lane + 32] * pw[lane + 32];
  for (int off = 16; off > 0; off >>= 1) d += __shfl_xor(d, off, 32);
  if (lane == 0) score[gw] = tanhf(d * (*invw));
}

// Per graph: keep top-K scores (stable lower-index tie-break), then
// pooled[g] = (1/K) * sum_{kept} score_i * x[i]   (order-invariant mean pool)
__global__ void topk_pool_kernel(const float* __restrict__ score, const float* __restrict__ X,
                                 float* __restrict__ pooled) {
  __shared__ float sv[NPG];
  __shared__ int   kp[NPG];
  int g = blockIdx.x;
  int t = threadIdx.x;
  if (t < NPG) sv[t] = score[g * NPG + t];
  __syncthreads();
  if (t < NPG) {
    float si = sv[t];
    int rank = 0;
    for (int j = 0; j < NPG; ++j) {
      float sj = sv[j];
      rank += (sj > si) || (sj == si && j < t);
    }
    kp[t] = (rank < KTOP) ? 1 : 0;
  }
  __syncthreads();
  if (t < DIM) {
    float a = 0.f;
    const float* xg = X + (size_t)g * NPG * DIM + t;
    for (int n = 0; n < NPG; ++n)
      if (kp[n]) a += sv[n] * xg[(size_t)n * DIM];
    pooled[g * DIM + t] = a * (1.0f / (float)KTOP);
  }
}

// logits = pooled @ W_lin + b ; out = log_softmax(logits)
__global__ void head_kernel(const float* __restrict__ pooled, const float* __restrict__ Wl,
                            const float* __restrict__ bl, float* __restrict__ out) {
  __shared__ float p[DIM];
  __shared__ float lg[NOUT];
  __shared__ float lse;
  int g = blockIdx.x, t = threadIdx.x;
  p[t] = pooled[g * DIM + t];
  __syncthreads();
  if (t < NOUT) {
    float l = bl[t];
    for (int d = 0; d < DIM; ++d) l += p[d] * Wl[d * NOUT + t];
    lg[t] = l;
  }
  __syncthreads();
  if (t == 0) {
    float mx = lg[0];
    for (int i = 1; i < NOUT; ++i) mx = fmaxf(mx, lg[i]);
    float s = 0.f;
    for (int i = 0; i < NOUT; ++i) s += expf(lg[i] - mx);
    lse = mx + logf(s);
  }
  __syncthreads();
  if (t < NOUT) out[g * NOUT + t] = lg[t] - lse;
}

// ---------------------------------------------------------------- launcher
extern "C" void kernel_launch(void* const* d_in, const int* in_sizes, int n_in,
                              void* d_out, int out_size, void* d_ws, size_t ws_size,
                              hipStream_t stream) {
  const float* x  = (const float*)d_in[0];
  const int* eidx = (const int*)d_in[1];
  // d_in[2] = batch (unused: graphs are contiguous, 500 nodes each)
  const float* W0 = (const float*)d_in[3];
  const float* b0 = (const float*)d_in[4];
  const float* W1 = (const float*)d_in[5];
  const float* b1 = (const float*)d_in[6];
  const float* pw = (const float*)d_in[7];
  const float* Wl = (const float*)d_in[8];
  const float* bl = (const float*)d_in[9];
  float* out = (float*)d_out;
  const int* row = eidx;            // edge_index[0] = sources
  const int* col = eidx + NEDGES;   // edge_index[1] = targets

  float* ws    = (float*)d_ws;
  float* bufA  = ws;                            // h  (50000*64)
  float* bufB  = bufA + (size_t)NNODES * DIM;   // acc / y (50000*64)
  float* dis   = bufB + (size_t)NNODES * DIM;   // 50000
  float* scr   = dis + NNODES;                  // deg, then score (50000)
  float* pooled = scr + NNODES;                 // 100*64
  float* invw  = pooled + NGRAPH * DIM;         // 1

  const int TPB = 256;
  const int fin_threads  = NNODES * DIM;        // 3.2M
  const int scat_threads = NEDGES * 16;         // 12.8M
  const int gemm_threads = (NNODES / 16) * 32;  // 3125 waves

  // degrees (with self loop) -> dis = deg^-1/2
  set_val_kernel<<<(NNODES + TPB - 1) / TPB, TPB, 0, stream>>>(scr, 1.0f, NNODES);
  deg_kernel<<<(NEDGES + TPB - 1) / TPB, TPB, 0, stream>>>(col, scr, NEDGES);
  rsqrt_kernel<<<(NNODES + TPB - 1) / TPB, TPB, 0, stream>>>(scr, dis, NNODES);

  // conv1: h = x@W0 ; acc = scatter ; y1 = relu(acc + self + b0) (in bufB)
  gemm_n64_wmma<<<(gemm_threads + TPB - 1) / TPB, TPB, 0, stream>>>(x, W0, bufA, NNODES);
  set_val_kernel<<<(fin_threads + TPB - 1) / TPB, TPB, 0, stream>>>(bufB, 0.0f, fin_threads);
  edge_scatter_kernel<<<(scat_threads + TPB - 1) / TPB, TPB, 0, stream>>>(row, col, dis, bufA, bufB, NEDGES);
  finalize_kernel<<<(fin_threads + TPB - 1) / TPB, TPB, 0, stream>>>(bufB, bufA, dis, b0, fin_threads);

  // conv2: h2 = y1@W1 (bufB -> bufA), then reuse bufB as acc/y2
  gemm_n64_wmma<<<(gemm_threads + TPB - 1) / TPB, TPB, 0, stream>>>(bufB, W1, bufA, NNODES);
  set_val_kernel<<<(fin_threads + TPB - 1) / TPB, TPB, 0, stream>>>(bufB, 0.0f, fin_threads);
  edge_scatter_kernel<<<(scat_threads + TPB - 1) / TPB, TPB, 0, stream>>>(row, col, dis, bufA, bufB, NEDGES);
  finalize_kernel<<<(fin_threads + TPB - 1) / TPB, TPB, 0, stream>>>(bufB, bufA, dis, b1, fin_threads);

  // TopK pooling + head
  wnorm_kernel<<<1, 64, 0, stream>>>(pw, invw);
  score_kernel<<<(NNODES * 32 + TPB - 1) / TPB, TPB, 0, stream>>>(bufB, pw, invw, scr, NNODES);
  topk_pool_kernel<<<NGRAPH, 512, 0, stream>>>(scr, bufB, pooled);
  head_kernel<<<NGRAPH, 64, 0, stream>>>(pooled, Wl, bl, out);
}